// PolymorphicNeuron_78623671321014
// MI455X (gfx1250) — compile-verified
//
#include <hip/hip_runtime.h>

typedef __attribute__((ext_vector_type(2))) float v2f;
typedef __attribute__((ext_vector_type(4))) float v4f;
typedef __attribute__((ext_vector_type(8))) float v8f;

#define B_ 512
#define D_ 1024
#define M_ 256
#define DT2_HALF 0.00005f   // 0.5 * DT^2, DT = 0.01

// ---------------------------------------------------------------------------
// K1: logits[b,m] = sum_d x[b,d] * w_sel[m,d] + b_sel[m]
// One wave computes a 16x16 tile with V_WMMA_F32_16X16X4_F32, K-loop over D.
// A lane layout (16x4 f32 A): row = b0 + (lane&15), K pair at 2*(lane>>4).
// B lane layout (4x16 f32 B): col = m0 + (lane&15), rows v + 2*(lane>>4).
// Since w_sel is [M,D] row-major, B[k][n] = w_sel[m0+n][kk+k] -> also a
// contiguous float2 per lane. Identical addressing for A and B.
// ---------------------------------------------------------------------------
__global__ __launch_bounds__(32) void k_gemm1(const float* __restrict__ x,
                                              const float* __restrict__ w,
                                              const float* __restrict__ bias,
                                              float* __restrict__ logits) {
  const int lane = threadIdx.x;
  const int half = lane >> 4, r = lane & 15;
  const int b0 = blockIdx.x * 16, m0 = blockIdx.y * 16;
  const float* ap = x + (size_t)(b0 + r) * D_ + 2 * half;
  const float* bp = w + (size_t)(m0 + r) * D_ + 2 * half;
  v8f c = {};
#pragma unroll 8
  for (int kk = 0; kk < D_; kk += 4) {
    v2f a = *(const v2f*)(ap + kk);
    v2f b = *(const v2f*)(bp + kk);
    c = __builtin_amdgcn_wmma_f32_16x16x4_f32(false, a, false, b, (short)0, c,
                                              false, false);
  }
  const float bv = bias[m0 + r];
#pragma unroll
  for (int i = 0; i < 8; ++i) {
    // C/D layout: VGPR i -> row b0 + i + 8*half, col m0 + r
    logits[(size_t)(b0 + i + 8 * half) * M_ + (m0 + r)] = c[i] + bv;
  }
}

// ---------------------------------------------------------------------------
// K2: in-place row softmax over M=256 columns. One wave per row, 8 cols/lane.
// ---------------------------------------------------------------------------
__global__ __launch_bounds__(32) void k_softmax(float* __restrict__ s) {
  const int b = blockIdx.x, lane = threadIdx.x;
  float v[8];
  float mx = -3.402823466e38f;
#pragma unroll
  for (int j = 0; j < 8; ++j) {
    v[j] = s[(size_t)b * M_ + j * 32 + lane];
    mx = fmaxf(mx, v[j]);
  }
#pragma unroll
  for (int off = 16; off; off >>= 1) mx = fmaxf(mx, __shfl_xor(mx, off, 32));
  float sum = 0.f;
#pragma unroll
  for (int j = 0; j < 8; ++j) {
    v[j] = __expf(v[j] - mx);
    sum += v[j];
  }
#pragma unroll
  for (int off = 16; off; off >>= 1) sum += __shfl_xor(sum, off, 32);
  const float inv = 1.0f / sum;
#pragma unroll
  for (int j = 0; j < 8; ++j) s[(size_t)b * M_ + j * 32 + lane] = v[j] * inv;
}

// ---------------------------------------------------------------------------
// K3: avg over batch per mode -> suppression scale (0.1 if sleeping, else 1).
// ---------------------------------------------------------------------------
__global__ __launch_bounds__(32) void k_avg(const float* __restrict__ s,
                                            const float* __restrict__ thr_p,
                                            float* __restrict__ scale) {
  const int m = blockIdx.x * 32 + threadIdx.x;
  float acc = 0.f;
  for (int b = 0; b < B_; ++b) acc += s[(size_t)b * M_ + m];
  const float avg = acc * (1.0f / B_);
  scale[m] = (avg > thr_p[0]) ? 0.1f : 1.0f;
}

// ---------------------------------------------------------------------------
// K4: equilibrium[b,d] = scores @ force_weights via WMMA f32 16x16x4, fused
// with spring dynamics. velocity == 0 so total_force = -k * (x - eq), and
// x_mod = x + 0.5*tf*dt^2. x_mod IS mem[0,:,:] so write it there directly.
// A = scores [B,M] (K contiguous, float2/lane). B = force_weights [M,D]
// (K is the row dim, stride D): two strided b32 loads per lane per K-step.
// ---------------------------------------------------------------------------
__global__ __launch_bounds__(32) void k_gemm2(const float* __restrict__ scores,
                                              const float* __restrict__ fw,
                                              const float* __restrict__ x,
                                              const float* __restrict__ spring_p,
                                              float* __restrict__ tf_out,
                                              float* __restrict__ xmod_out) {
  const int lane = threadIdx.x;
  const int half = lane >> 4, r = lane & 15;
  const int b0 = blockIdx.x * 16, d0 = blockIdx.y * 16;
  const float* ap = scores + (size_t)(b0 + r) * M_ + 2 * half;
  const float* bp = fw + (size_t)(2 * half) * D_ + d0 + r;
  v8f c = {};
#pragma unroll 4
  for (int kk = 0; kk < M_; kk += 4) {
    v2f a = *(const v2f*)(ap + kk);
    v2f b;
    b.x = bp[(size_t)kk * D_];        // row kk + 2*half
    b.y = bp[(size_t)kk * D_ + D_];   // row kk + 2*half + 1
    c = __builtin_amdgcn_wmma_f32_16x16x4_f32(false, a, false, b, (short)0, c,
                                              false, false);
  }
  const float ks = spring_p[0];
#pragma unroll
  for (int i = 0; i < 8; ++i) {
    const size_t idx = (size_t)(b0 + i + 8 * half) * D_ + d0 + r;
    const float xv = x[idx];
    const float tf = -ks * (xv - c[i]);
    tf_out[idx] = tf;
    xmod_out[idx] = xv + tf * DT2_HALF;
  }
}

// ---------------------------------------------------------------------------
// K5: scores_sup = scores * scale[m] in place; S[b] = row sum (used because
// spk is m-independent: mixed = H(x_mod-1) * S[b]).
// ---------------------------------------------------------------------------
__global__ __launch_bounds__(32) void k_rowscale(float* __restrict__ s,
                                                 const float* __restrict__ scale,
                                                 float* __restrict__ S) {
  const int b = blockIdx.x, lane = threadIdx.x;
  float sum = 0.f;
#pragma unroll
  for (int j = 0; j < 8; ++j) {
    const int m = j * 32 + lane;
    const float v = s[(size_t)b * M_ + m] * scale[m];
    s[(size_t)b * M_ + m] = v;
    sum += v;
  }
#pragma unroll
  for (int off = 16; off; off >>= 1) sum += __shfl_xor(sum, off, 32);
  if (lane == 0) S[b] = sum;
}

// ---------------------------------------------------------------------------
// K6: the bandwidth kernel. Read x_mod (mem slice 0) once as float4, emit
// mixed = H(x_mod - 1) * S[b], then broadcast 255 more mem slices with
// non-temporal b128 stores (536 MB total; never re-read -> NT is correct).
// ---------------------------------------------------------------------------
__global__ __launch_bounds__(256) void k_broadcast(const float* __restrict__ xmod,
                                                   const float* __restrict__ S,
                                                   float* __restrict__ mem,
                                                   float* __restrict__ mixed) {
  const size_t idx = (size_t)blockIdx.x * 256 + threadIdx.x;  // float4 index
  const v4f v = ((const v4f*)xmod)[idx];
  const int b = (int)(idx >> 8);  // (idx*4)/D_
  const float Sb = S[b];
  v4f mix;
  mix.x = (v.x > 1.0f) ? Sb : 0.0f;
  mix.y = (v.y > 1.0f) ? Sb : 0.0f;
  mix.z = (v.z > 1.0f) ? Sb : 0.0f;
  mix.w = (v.w > 1.0f) ? Sb : 0.0f;
  ((v4f*)mixed)[idx] = mix;
#pragma unroll 4
  for (int m = 1; m < M_; ++m) {
    __builtin_nontemporal_store(v, (v4f*)(mem + (size_t)m * (B_ * D_)) + idx);
  }
}

// ---------------------------------------------------------------------------
extern "C" void kernel_launch(void* const* d_in, const int* in_sizes, int n_in,
                              void* d_out, int out_size, void* d_ws,
                              size_t ws_size, hipStream_t stream) {
  (void)in_sizes; (void)n_in; (void)out_size; (void)ws_size;
  const float* x      = (const float*)d_in[0];
  const float* w_sel  = (const float*)d_in[1];
  const float* b_sel  = (const float*)d_in[2];
  const float* fw     = (const float*)d_in[3];
  // d_in[4] damping_factor unused: velocity == 0 on first step
  const float* spring = (const float*)d_in[5];
  const float* thr    = (const float*)d_in[6];

  float* out    = (float*)d_out;
  float* mixed  = out;                                  // [B,D]
  float* mem    = out + (size_t)B_ * D_;                // [M,B,D]
  float* scores = mem + (size_t)M_ * B_ * D_;           // [B,M] (scores_sup)
  float* tf     = scores + (size_t)B_ * M_;             // [B,D] (total_force)

  float* scale = (float*)d_ws;   // [M_]
  float* S     = scale + M_;     // [B_]

  k_gemm1<<<dim3(B_ / 16, M_ / 16), 32, 0, stream>>>(x, w_sel, b_sel, scores);
  k_softmax<<<B_, 32, 0, stream>>>(scores);
  k_avg<<<M_ / 32, 32, 0, stream>>>(scores, thr, scale);
  // GEMM2 must read UNSCALED scores -> run before k_rowscale mutates them.
  k_gemm2<<<dim3(B_ / 16, D_ / 16), 32, 0, stream>>>(scores, fw, x, spring, tf,
                                                     mem /* slice 0 = x_mod */);
  k_rowscale<<<B_, 32, 0, stream>>>(scores, scale, S);
  k_broadcast<<<(B_ * D_ / 4) / 256, 256, 0, stream>>>(mem, S, mem, mixed);
}